// Fixation_48619029791083
// MI455X (gfx1250) — compile-verified
//
#include <hip/hip_runtime.h>

typedef float v2f   __attribute__((ext_vector_type(2)));
typedef float v8f   __attribute__((ext_vector_type(8)));
typedef float f32x4 __attribute__((ext_vector_type(4)));

#define SEQ     197
#define SS      (SEQ * SEQ)     // 38809
#define NHEADS  12
#define NP      196             // patches
#define CUT     98              // top-k cutoff
#define STRIDE  224             // padded row stride in workspace (>= 208)
#define IMG_ELEMS 150528        // 3*224*224
#define CH_ELEMS  50176         // 224*224

// ---------------------------------------------------------------------------
// Kernel 1: attn[b,p] = sum_h x[b,h,0,1+p] via V_WMMA_F32_16X16X4_F32.
// One wave per 16-patch tile. A(16x4) = attention values (M=patch, K=head),
// B(4x16) = all ones (uniform, so B's VGPR layout is irrelevant),
// D[m][n] = sum_k A[m][k] for every n. K=12 accumulated as 3 chunks of 4.
// A layout (ISA 7.12.2, 32-bit A 16x4): lanes 0-15 -> K=0,1; lanes 16-31 -> K=2,3.
// D layout: lanes 0-15 VGPR r = row r; lanes 16-31 VGPR r = row r+8.
// Store tail is unguarded: rows land in the stride-224 padded workspace row
// (max index 207 < 224), emitted as two aligned global_store_b128.
// ---------------------------------------------------------------------------
__global__ __launch_bounds__(32) void headsum_wmma(const float* __restrict__ x,
                                                   float* __restrict__ attn) {
    const int b    = blockIdx.y;
    const int t    = blockIdx.x;            // 0..12  (13 tiles cover 196+pad)
    const int lane = threadIdx.x;           // 0..31
    const int m    = lane & 15;             // patch-row within tile
    int p = t * 16 + m;
    int p_eff = p < NP ? p : (NP - 1);      // clamp pad rows (never consumed)
    const int kpair = (lane >> 4) * 2;      // 0 or 2

    const float* xb = x + (size_t)b * NHEADS * SS + 1 + p_eff;

    v8f c = {};
    v2f ones; ones[0] = 1.0f; ones[1] = 1.0f;

#pragma unroll
    for (int chunk = 0; chunk < 3; ++chunk) {
        const int k0 = chunk * 4 + kpair;
        v2f a;
        a[0] = xb[(size_t)(k0)     * SS];
        a[1] = xb[(size_t)(k0 + 1) * SS];
        c = __builtin_amdgcn_wmma_f32_16x16x4_f32(
                /*neg_a=*/false, a, /*neg_b=*/false, ones,
                /*c_mod=*/(short)0, c, /*reuse_a=*/false, /*reuse_b=*/false);
    }

    // Lane 0 holds rows 0..7 of the tile; lane 16 holds rows 8..15.
    if ((lane & 15) == 0) {
        const int pbase = t * 16 + (lane >> 4) * 8;   // multiple of 8 -> 32B aligned
        f32x4 lo; lo[0] = c[0]; lo[1] = c[1]; lo[2] = c[2]; lo[3] = c[3];
        f32x4 hi; hi[0] = c[4]; hi[1] = c[5]; hi[2] = c[6]; hi[3] = c[7];
        f32x4* dst = (f32x4*)(attn + b * STRIDE + pbase);
        dst[0] = lo;
        dst[1] = hi;
    }
}

// ---------------------------------------------------------------------------
// Kernel 2: exact top-98 rank select per batch (matches jax.lax.top_k tie
// rule: ties broken toward smaller index). mask[p] = 1 iff rank(p) < 98.
// ---------------------------------------------------------------------------
__global__ __launch_bounds__(256) void rank_mask(const float* __restrict__ attn,
                                                 float* __restrict__ mask) {
    __shared__ float s[NP];
    const int b   = blockIdx.x;
    const int tid = threadIdx.x;
    if (tid < NP) s[tid] = attn[b * STRIDE + tid];
    __syncthreads();
    if (tid < NP) {
        const float a = s[tid];
        int cnt = 0;
#pragma unroll 4
        for (int j = 0; j < NP; ++j) {
            const float aj = s[j];
            cnt += (aj > a) || (aj == a && j < tid);
        }
        mask[b * STRIDE + tid] = (cnt < CUT) ? 1.0f : 0.0f;
    }
}

// ---------------------------------------------------------------------------
// Kernel 3: streaming masked multiply, float4-vectorized, non-temporal on the
// 154MB+154MB image streams (they exceed 192MB L2; mask stays cached).
// A 4-aligned float4 never crosses a 16-pixel patch boundary -> 1 mask/vec.
// ---------------------------------------------------------------------------
__global__ __launch_bounds__(256) void apply_mask(const float* __restrict__ img,
                                                  const float* __restrict__ mask,
                                                  float* __restrict__ out) {
    const int b  = blockIdx.y;
    const int j4 = blockIdx.x * 256 + threadIdx.x;  // 0..37631
    const int i  = j4 * 4;                          // element offset in image
    const int p  = i % CH_ELEMS;                    // within channel plane
    const int y  = p / 224;
    const int xx = p % 224;
    const int patch = (y >> 4) * 14 + (xx >> 4);

    const float m = mask[b * STRIDE + patch];
    const size_t off = (size_t)b * IMG_ELEMS + (size_t)i;

    f32x4 v = __builtin_nontemporal_load((const f32x4*)(img + off));
    f32x4 r = v * m;
    __builtin_nontemporal_store(r, (f32x4*)(out + off));
}

// ---------------------------------------------------------------------------
extern "C" void kernel_launch(void* const* d_in, const int* in_sizes, int n_in,
                              void* d_out, int out_size, void* d_ws, size_t ws_size,
                              hipStream_t stream) {
    const float* x   = (const float*)d_in[0];   // [256,12,197,197] f32
    const float* img = (const float*)d_in[1];   // [256,3,224,224]  f32
    float* out = (float*)d_out;                 // [256, 150528]    f32

    float* attn = (float*)d_ws;                 // 256*224 floats
    float* mask = attn + 256 * STRIDE;          // 256*224 floats

    headsum_wmma<<<dim3(13, 256), 32, 0, stream>>>(x, attn);
    rank_mask  <<<256, 256, 0, stream>>>(attn, mask);
    apply_mask <<<dim3(147, 256), 256, 0, stream>>>(img, mask, out);
}